// SequentialGraphFeaturesExtractor_18665927868854
// MI455X (gfx1250) — compile-verified
//
#include <hip/hip_runtime.h>

typedef float v2f __attribute__((ext_vector_type(2)));
typedef float v8f __attribute__((ext_vector_type(8)));

#define NEG_SLOPE 0.2f
#define EPS_F 1e-16f

// ---- ordered-int encoding so signed-int atomicMax == float max -------------
__device__ __forceinline__ int orderedF(float f) {
  int i = __float_as_int(f);
  return (i >= 0) ? i : (i ^ 0x7FFFFFFF);
}
__device__ __forceinline__ float unorderedF(int i) {
  return __int_as_float((i >= 0) ? i : (i ^ 0x7FFFFFFF));
}

__device__ __forceinline__ void get_edge(const long long* __restrict__ ei,
                                         int e, int E, int* s, int* d) {
  if (e < E) { *s = (int)ei[e]; *d = (int)ei[(size_t)E + e]; }
  else       { *s = e - E; *d = e - E; }   // self loop
}

// ---------------------------------------------------------------------------
// Fill kernels (graph-capture safe initialization of workspace)
// ---------------------------------------------------------------------------
__global__ void fill_f32(float* __restrict__ p, float v, int n) {
  int i = blockIdx.x * blockDim.x + threadIdx.x;
  if (i < n) p[i] = v;
}
__global__ void fill_i32(int* __restrict__ p, int v, int n) {
  int i = blockIdx.x * blockDim.x + threadIdx.x;
  if (i < n) p[i] = v;
}

// ---------------------------------------------------------------------------
// Exact-fp32 WMMA GEMM with B staged in LDS as K-interleaved pairs.
//   C[M x Nc] = A[M x K] @ B[K x Nc] (+bias)(+relu), all row-major.
//
// Block = 256 threads = 8 waves; a block owns 8 consecutive M-tiles and TW
// consecutive N-tiles (TW*16 columns). B[:, colBase:colBase+TW*16] is loaded
// once into LDS in float2-pair form:
//     sB2[p * S2 + c] = ( B[2p][c], B[2p+1][c] )
// so each lane's V_WMMA_F32_16X16X4_F32 B fragment is ONE aligned ds_load_b64
// (lane-half 0 -> pair-row k/2 = K0/K1, half 1 -> pair-row k/2+1 = K2/K3),
// matching the ISA 4x16 fragment striping with zero register shuffling.
// S2 = TWC+16 makes 2*S2 = 32 (mod 64 banks): the two lane-halves hit
// disjoint 32-bank sets -> conflict-free 8B accesses.
//
// Per K-step per wave: 1 global_load_b64 (A pair, 8B-aligned since K%4==0)
// + TW ds_load_b64 + TW WMMAs (A fragment reused across all N-tiles).
// ---------------------------------------------------------------------------
template <int TW>
__global__ void __launch_bounds__(256)
wmma_gemm_f32_lds(const float* __restrict__ A, const float* __restrict__ B,
                  const float* __restrict__ bias, float* __restrict__ C,
                  int M, int K, int Nc, int relu) {
  constexpr int TWC = TW * 16;       // columns handled per block
  constexpr int S2  = TWC + 16;      // pair-row stride in float2 units
  __shared__ v2f sB2[64 * (64 + 16)];   // worst case K/2=64 pair-rows, 40 KB

  const int nGroups = Nc / TWC;
  const int mTiles  = M >> 4;
  const int mBlock  = blockIdx.x / nGroups;
  const int nGroup  = blockIdx.x - mBlock * nGroups;
  const int colBase = nGroup * TWC;

  // Cooperative pair-interleaved load of B[:, colBase:colBase+TWC] into LDS.
  // Reached unconditionally so the barrier is safe for partial M-blocks.
  for (int idx = threadIdx.x; idx < (K / 2) * TWC; idx += 256) {
    const int p = idx / TWC;
    const int c = idx - p * TWC;
    v2f t2;
    t2.x = B[(size_t)(2 * p)     * Nc + colBase + c];
    t2.y = B[(size_t)(2 * p + 1) * Nc + colBase + c];
    sB2[p * S2 + c] = t2;
  }
  __syncthreads();

  const int mTile = mBlock * 8 + (threadIdx.x >> 5);
  if (mTile >= mTiles) return;       // wave-uniform, after the barrier

  const int lane = threadIdx.x & 31;
  const int half = lane >> 4;        // 0: lanes 0-15 (K0/K1), 1: lanes 16-31 (K2/K3)
  const int l16  = lane & 15;
  const int rowA = mTile * 16 + l16;

  v8f acc[TW];
#pragma unroll
  for (int t = 0; t < TW; ++t)
#pragma unroll
    for (int i = 0; i < 8; ++i) acc[t][i] = 0.0f;

  for (int k = 0; k < K; k += 4) {
    const v2f a = *(const v2f*)&A[(size_t)rowA * K + k + 2 * half];
    const int pr = (k >> 1) + half;  // pair-row for this lane-half
#pragma unroll
    for (int t = 0; t < TW; ++t) {
      const v2f b = sB2[pr * S2 + t * 16 + l16];   // single ds_load_b64
      acc[t] = __builtin_amdgcn_wmma_f32_16x16x4_f32(
          /*neg_a=*/false, a, /*neg_b=*/false, b,
          /*c_mod=*/(short)0, acc[t], /*reuse_a=*/false, /*reuse_b=*/false);
    }
  }

  const int rowBase = mTile * 16 + half * 8;
#pragma unroll
  for (int t = 0; t < TW; ++t) {
    const int col = colBase + t * 16 + l16;
    const float bv = bias ? bias[col] : 0.0f;
#pragma unroll
    for (int i = 0; i < 8; ++i) {
      float v = acc[t][i] + bv;
      if (relu) v = fmaxf(v, 0.0f);
      C[(size_t)(rowBase + i) * Nc + col] = v;
    }
  }
}

// ---------------------------------------------------------------------------
// Per-node attention terms: alpha_s[n] = <h[n], a_src>, alpha_d[n] = <h[n], a_dst>
// ---------------------------------------------------------------------------
__global__ void node_attn(const float* __restrict__ h, const float* __restrict__ a_s,
                          const float* __restrict__ a_d, float* __restrict__ out_s,
                          float* __restrict__ out_d, int N, int C) {
  int n = blockIdx.x * blockDim.x + threadIdx.x;
  if (n >= N) return;
  const float* hp = h + (size_t)n * C;
  float ss = 0.0f, sd = 0.0f;
  for (int c = 0; c < C; ++c) {
    float v = hp[c];
    ss = fmaf(v, a_s[c], ss);
    sd = fmaf(v, a_d[c], sd);
  }
  out_s[n] = ss;
  out_d[n] = sd;
}

// ---------------------------------------------------------------------------
// Segment-softmax over edges grouped by dst
// ---------------------------------------------------------------------------
__global__ void edge_max(const long long* __restrict__ ei, int E, int N,
                         const float* __restrict__ as, const float* __restrict__ ad,
                         int* __restrict__ mmax) {
  int e = blockIdx.x * blockDim.x + threadIdx.x;
  if (e >= E + N) return;
  int s, d; get_edge(ei, e, E, &s, &d);
  float x = as[s] + ad[d];
  x = (x > 0.0f) ? x : x * NEG_SLOPE;             // LeakyReLU(0.2)
  atomicMax(&mmax[d], orderedF(x));
}

__global__ void edge_exp(const long long* __restrict__ ei, int E, int N,
                         const float* __restrict__ as, const float* __restrict__ ad,
                         const int* __restrict__ mmax, float* __restrict__ eexp,
                         float* __restrict__ denom) {
  int e = blockIdx.x * blockDim.x + threadIdx.x;
  if (e >= E + N) return;
  int s, d; get_edge(ei, e, E, &s, &d);
  float x = as[s] + ad[d];
  x = (x > 0.0f) ? x : x * NEG_SLOPE;
  float ex = expf(x - unorderedF(mmax[d]));
  eexp[e] = ex;
  atomicAdd(&denom[d], ex);
}

// Warp-per-edge weighted scatter: out[dst] += alpha * h[src]
__global__ void __launch_bounds__(256)
edge_scatter(const long long* __restrict__ ei, int E, int N,
             const float* __restrict__ h, const float* __restrict__ eexp,
             const float* __restrict__ denom, float* __restrict__ out, int C) {
  int e = blockIdx.x * (blockDim.x >> 5) + (threadIdx.x >> 5);
  if (e >= E + N) return;
  int lane = threadIdx.x & 31;
  int s, d; get_edge(ei, e, E, &s, &d);
  const float* hp = h + (size_t)s * C;
  // Kick the random-gather cacheline fetch early; overlaps with the division.
  __builtin_prefetch(hp + lane, 0, 3);
  float alpha = eexp[e] / (denom[d] + EPS_F);
  float* op = out + (size_t)d * C;
  for (int c = lane; c < C; c += 32)
    atomicAdd(&op[c], alpha * hp[c]);
}

__global__ void bias_relu(float* __restrict__ h, const float* __restrict__ bias,
                          int N, int C) {
  int i = blockIdx.x * blockDim.x + threadIdx.x;
  if (i >= N * C) return;
  h[i] = fmaxf(h[i] + bias[i % C], 0.0f);
}

// ---------------------------------------------------------------------------
// Host-side orchestration
// ---------------------------------------------------------------------------
static inline int cdiv(int a, int b) { return (a + b - 1) / b; }

extern "C" void kernel_launch(void* const* d_in, const int* in_sizes, int n_in,
                              void* d_out, int out_size, void* d_ws, size_t ws_size,
                              hipStream_t stream) {
  const float*     x   = (const float*)d_in[0];
  const long long* ei  = (const long long*)d_in[1];   // int64 edge_index (2,E)
  // d_in[2] = edge_attr: ignored (GATConv with edge_dim=None)
  const float* W1 = (const float*)d_in[3];
  const float* as1 = (const float*)d_in[4];
  const float* ad1 = (const float*)d_in[5];
  const float* b1 = (const float*)d_in[6];
  const float* W2 = (const float*)d_in[7];
  const float* as2 = (const float*)d_in[8];
  const float* ad2 = (const float*)d_in[9];
  const float* b2 = (const float*)d_in[10];
  const float* W3 = (const float*)d_in[11];
  const float* as3 = (const float*)d_in[12];
  const float* ad3 = (const float*)d_in[13];
  const float* b3 = (const float*)d_in[14];
  const float* Wl = (const float*)d_in[15];
  const float* bl = (const float*)d_in[16];

  const int N  = in_sizes[0] / 128;   // 100000 (divisible by 16: 6250 M-tiles)
  const int E  = in_sizes[1] / 2;     // 1600000
  const int EN = E + N;               // edges incl. self loops

  // Workspace layout (floats): ~60 MB total
  float* bufA  = (float*)d_ws;                 // N*64  (post-GEMM features h)
  float* bufB  = bufA + (size_t)N * 64;        // N*64  (aggregated output / next input)
  float* aS    = bufB + (size_t)N * 64;        // N
  float* aD    = aS + N;                       // N
  float* denom = aD + N;                       // N
  int*   mmax  = (int*)(denom + N);            // N
  float* eexp  = (float*)(mmax + N);           // E+N

  const int mBlocks = cdiv(N / 16, 8);

  auto gemm = [&](const float* A, const float* B, const float* bias, float* C,
                  int K, int Nc, int relu) {
    if (Nc >= 64) {
      wmma_gemm_f32_lds<4><<<mBlocks * (Nc / 64), 256, 0, stream>>>(
          A, B, bias, C, N, K, Nc, relu);
    } else {  // Nc == 32
      wmma_gemm_f32_lds<2><<<mBlocks, 256, 0, stream>>>(
          A, B, bias, C, N, K, Nc, relu);
    }
  };

  auto gat_layer = [&](const float* h_in, int Cin, int Cout,
                       const float* W, const float* a_s, const float* a_d,
                       const float* bias) {
    // h = h_in @ W  (exact fp32 WMMA, B staged in LDS)
    gemm(h_in, W, nullptr, bufA, Cin, Cout, 0);
    // per-node attention scalars
    node_attn<<<cdiv(N, 256), 256, 0, stream>>>(bufA, a_s, a_d, aS, aD, N, Cout);
    // init segment buffers + output accumulator
    fill_i32<<<cdiv(N, 256), 256, 0, stream>>>(mmax, (int)0x80000000, N);
    fill_f32<<<cdiv(N, 256), 256, 0, stream>>>(denom, 0.0f, N);
    fill_f32<<<cdiv(N * Cout, 256), 256, 0, stream>>>(bufB, 0.0f, N * Cout);
    // segment softmax over dst + weighted scatter
    edge_max<<<cdiv(EN, 256), 256, 0, stream>>>(ei, E, N, aS, aD, mmax);
    edge_exp<<<cdiv(EN, 256), 256, 0, stream>>>(ei, E, N, aS, aD, mmax, eexp, denom);
    edge_scatter<<<cdiv(EN, 8), 256, 0, stream>>>(ei, E, N, bufA, eexp, denom,
                                                  bufB, Cout);
    // bias + ReLU (heads=1 so concat/mean are both identity)
    bias_relu<<<cdiv(N * Cout, 256), 256, 0, stream>>>(bufB, bias, N, Cout);
  };

  gat_layer(x,    128, 32, W1, as1, ad1, b1);   // layer 1: 128 -> 32
  gat_layer(bufB,  32, 64, W2, as2, ad2, b2);   // layer 2: 32 -> 64
  gat_layer(bufB,  64, 64, W3, as3, ad3, b3);   // layer 3: 64 -> 64

  // Final linear 64 -> 512 with fused bias, straight to d_out (fp32)
  gemm(bufB, Wl, bl, (float*)d_out, 64, 512, 0);
}